// DynamicTopologicalLayer_3934190043625
// MI455X (gfx1250) — compile-verified
//
#include <hip/hip_runtime.h>
#include <math.h>

typedef float v8f __attribute__((ext_vector_type(8)));
typedef float v2f __attribute__((ext_vector_type(2)));

#define HID 128
#define HEADS 4
#define HEAD_DIM 32
#define NEG_SLOPE 0.2f
#define LN_EPS 1e-5f

// ---------------------------------------------------------------------------
// helpers
// ---------------------------------------------------------------------------
__device__ __forceinline__ float atomic_add_f32(float* addr, float v) {
    return __hip_atomic_fetch_add(addr, v, __ATOMIC_RELAXED, __HIP_MEMORY_SCOPE_AGENT);
}

// float atomic max via signed/unsigned int monotonic-bits trick
__device__ __forceinline__ void atomic_max_f32(float* addr, float v) {
    if (v >= 0.0f) {
        atomicMax((int*)addr, __float_as_int(v));
    } else {
        atomicMin((unsigned int*)addr, (unsigned int)__float_as_int(v));
    }
}

__device__ __forceinline__ float leaky(float x) {
    return x > 0.0f ? x : NEG_SLOPE * x;
}

// One 16x16 (M x N) fp32 WMMA tile with K=128, A from row-major [16][128] buffer,
// B from row-major [128+][HID] global weight (column slice `col`), scaled by bscale.
// Layout per CDNA5 ISA 7.12.2:
//   A 16x4: lanes 0-15 -> M=lane, VGPR0=K0,VGPR1=K1 ; lanes 16-31 -> K2/K3
//   C/D  : VGPR v -> M=v (lanes 0-15) / M=v+8 (lanes 16-31), N=lane&15
__device__ __forceinline__ v8f wmma_tile_k128(const float* __restrict__ abuf,
                                              const float* __restrict__ bptr,
                                              float bscale, int m, int kb, int col,
                                              v8f c) {
    for (int k0 = 0; k0 < 128; k0 += 4) {
        v2f a, b;
        a.x = abuf[m * HID + k0 + kb];
        a.y = abuf[m * HID + k0 + kb + 1];
        b.x = bptr[(k0 + kb) * HID + col] * bscale;
        b.y = bptr[(k0 + kb + 1) * HID + col] * bscale;
        c = __builtin_amdgcn_wmma_f32_16x16x4_f32(false, a, false, b, (short)0, c,
                                                  false, false);
    }
    return c;
}

// Per-wave LayerNorm over 2 rows of a [16][128] LDS buffer (in place), optional ReLU.
__device__ __forceinline__ void ln_rows_inplace(float* buf, const float* __restrict__ g,
                                                const float* __restrict__ bb, int wave,
                                                int lane, bool relu) {
    for (int rr = 0; rr < 2; rr++) {
        int r = wave * 2 + rr;
        float s = 0.0f;
        for (int c = lane; c < HID; c += 32) s += buf[r * HID + c];
        for (int o = 16; o > 0; o >>= 1) s += __shfl_down(s, o, 32);
        s = __shfl(s, 0, 32);
        float mean = s * (1.0f / HID);
        float vs = 0.0f;
        for (int c = lane; c < HID; c += 32) {
            float d = buf[r * HID + c] - mean;
            vs += d * d;
        }
        for (int o = 16; o > 0; o >>= 1) vs += __shfl_down(vs, o, 32);
        vs = __shfl(vs, 0, 32);
        float inv = rsqrtf(vs * (1.0f / HID) + LN_EPS);
        for (int c = lane; c < HID; c += 32) {
            float v = (buf[r * HID + c] - mean) * inv * g[c] + bb[c];
            if (relu) v = v > 0.0f ? v : 0.0f;
            buf[r * HID + c] = v;
        }
    }
}

// ---------------------------------------------------------------------------
// kernel 1: h0 = x @ gat_w   (N x 128) = (N x 128)(128 x 128), WMMA f32
// grid: ceil(N/16) blocks of 256 threads (8 waves; wave w owns cols [16w,16w+16))
// ---------------------------------------------------------------------------
__global__ void k_gemm_xw(const float* __restrict__ x, const float* __restrict__ w,
                          float* __restrict__ h0, int n) {
    __shared__ float sx[16 * HID];
    int r0 = blockIdx.x * 16;
    int tid = threadIdx.x;
    for (int i = tid; i < 16 * HID; i += 256) {
        int r = i >> 7, c = i & 127;
        int gr = r0 + r;
        sx[i] = (gr < n) ? x[(long long)gr * HID + c] : 0.0f;
    }
    __syncthreads();
    int wave = tid >> 5, lane = tid & 31;
    int m = lane & 15;
    int kb = (lane >> 4) << 1;
    int col = wave * 16 + m;
    v8f c = {};
    c = wmma_tile_k128(sx, w, 1.0f, m, kb, col, c);
    int rb = r0 + ((lane >> 4) << 3);
    for (int v = 0; v < 8; v++) {
        int gr = rb + v;
        if (gr < n) h0[(long long)gr * HID + col] = c[v];
    }
}

// ---------------------------------------------------------------------------
// kernel 2: per (node, head) attention logits a_src / a_dst
// ---------------------------------------------------------------------------
__global__ void k_att_logits(const float* __restrict__ h0,
                             const float* __restrict__ att_src,
                             const float* __restrict__ att_dst,
                             float* __restrict__ asrc, float* __restrict__ adst,
                             int n) {
    int i = blockIdx.x * blockDim.x + threadIdx.x;
    if (i >= n * HEADS) return;
    int node = i >> 2, h = i & 3;
    const float* xp = h0 + (long long)node * HID + h * HEAD_DIM;
    const float* as = att_src + h * HEAD_DIM;
    const float* ad = att_dst + h * HEAD_DIM;
    float s = 0.0f, d = 0.0f;
    for (int c = 0; c < HEAD_DIM; c++) {
        float v = xp[c];
        s += v * as[c];
        d += v * ad[c];
    }
    asrc[i] = s;
    adst[i] = d;
}

// ---------------------------------------------------------------------------
// kernel 3: init reductions buffers
// ---------------------------------------------------------------------------
__global__ void k_init(float* __restrict__ mbuf, float* __restrict__ denom,
                       float* __restrict__ agg, int n) {
    int i = blockIdx.x * blockDim.x + threadIdx.x;
    int gs = gridDim.x * blockDim.x;
    for (int j = i; j < n * HEADS; j += gs) {
        mbuf[j] = -__builtin_inff();
        denom[j] = 0.0f;
    }
    for (long long j = i; j < (long long)n * HID; j += gs) agg[j] = 0.0f;
}

// ---------------------------------------------------------------------------
// kernel 4: segment max of leaky-relu logits over destination nodes
// edges [0,E) from edge_index, [E, E+N) are self loops
// ---------------------------------------------------------------------------
__global__ void k_edge_max(const int* __restrict__ ei, const float* __restrict__ asrc,
                           const float* __restrict__ adst, float* __restrict__ mbuf,
                           int E, int n) {
    long long i = (long long)blockIdx.x * blockDim.x + threadIdx.x;
    long long tot = (long long)(E + n) * HEADS;
    if (i >= tot) return;
    int e = (int)(i >> 2), h = (int)(i & 3);
    int s, d;
    if (e < E) { s = ei[e]; d = ei[E + e]; } else { s = e - E; d = s; }
    float v = leaky(asrc[s * HEADS + h] + adst[d * HEADS + h]);
    atomic_max_f32(&mbuf[d * HEADS + h], v);
}

// ---------------------------------------------------------------------------
// kernel 5: e_exp and segment sum (softmax denominator)
// ---------------------------------------------------------------------------
__global__ void k_edge_exp(const int* __restrict__ ei, const float* __restrict__ asrc,
                           const float* __restrict__ adst,
                           const float* __restrict__ mbuf, float* __restrict__ eexp,
                           float* __restrict__ denom, int E, int n) {
    long long i = (long long)blockIdx.x * blockDim.x + threadIdx.x;
    long long tot = (long long)(E + n) * HEADS;
    if (i >= tot) return;
    int e = (int)(i >> 2), h = (int)(i & 3);
    int s, d;
    if (e < E) { s = ei[e]; d = ei[E + e]; } else { s = e - E; d = s; }
    float v = leaky(asrc[s * HEADS + h] + adst[d * HEADS + h]);
    float ee = __expf(v - mbuf[d * HEADS + h]);
    eexp[i] = ee;
    atomic_add_f32(&denom[d * HEADS + h], ee);
}

// ---------------------------------------------------------------------------
// kernel 6: weighted message scatter, one wave (32 lanes) per edge
// ---------------------------------------------------------------------------
__global__ void k_message(const int* __restrict__ ei, const float* __restrict__ h0,
                          const float* __restrict__ eexp,
                          const float* __restrict__ denom, float* __restrict__ agg,
                          int E, int n) {
    long long gt = (long long)blockIdx.x * blockDim.x + threadIdx.x;
    int e = (int)(gt >> 5);
    int lane = (int)(gt & 31);
    if (e >= E + n) return;
    int s, d;
    if (e < E) { s = ei[e]; d = ei[E + e]; } else { s = e - E; d = s; }
    const float* xs = h0 + (long long)s * HID;
    float* od = agg + (long long)d * HID;
    for (int h = 0; h < HEADS; h++) {
        float alpha = eexp[(long long)e * HEADS + h] /
                      (denom[d * HEADS + h] + 1e-16f);
        float val = xs[h * HEAD_DIM + lane] * alpha;
        atomic_add_f32(&od[h * HEAD_DIM + lane], val);
    }
}

// ---------------------------------------------------------------------------
// kernel 7: spectral norm of pc_w (128x128) by power iteration; writes 1/sigma
// single block of 128 threads
// ---------------------------------------------------------------------------
__global__ void k_sigma(const float* __restrict__ w, float* __restrict__ invsig) {
    __shared__ float v[HID], u[HID], red[4];
    int t = threadIdx.x;
    v[t] = 1.0f;
    __syncthreads();
    for (int it = 0; it < 40; it++) {
        float s = 0.0f;  // u = W v
        for (int j = 0; j < HID; j++) s += w[t * HID + j] * v[j];
        u[t] = s;
        __syncthreads();
        float s2 = 0.0f;  // v' = W^T u
        for (int i = 0; i < HID; i++) s2 += w[i * HID + t] * u[i];
        float sq = s2 * s2;
        for (int o = 16; o > 0; o >>= 1) sq += __shfl_down(sq, o, 32);
        if ((t & 31) == 0) red[t >> 5] = sq;
        __syncthreads();
        float nrm = sqrtf(red[0] + red[1] + red[2] + red[3]) + 1e-30f;
        v[t] = s2 / nrm;
        __syncthreads();
    }
    float s = 0.0f;  // sigma = ||W v||
    for (int j = 0; j < HID; j++) s += w[t * HID + j] * v[j];
    float sq = s * s;
    for (int o = 16; o > 0; o >>= 1) sq += __shfl_down(sq, o, 32);
    if ((t & 31) == 0) red[t >> 5] = sq;
    __syncthreads();
    if (t == 0) invsig[0] = 1.0f / (sqrtf(red[0] + red[1] + red[2] + red[3]) + 1e-30f);
}

// ---------------------------------------------------------------------------
// kernel 8: fused dense pipeline per 16-node tile (4 WMMA GEMM stages + 3 LNs)
// ---------------------------------------------------------------------------
__global__ void k_fused(const float* __restrict__ agg, const float* __restrict__ gat_b,
                        const float* __restrict__ pc_w, const float* __restrict__ pc_b,
                        const float* __restrict__ pc_g, const float* __restrict__ pc_bb,
                        const float* __restrict__ mgf_w, const float* __restrict__ mgf_b,
                        const float* __restrict__ sym_w, const float* __restrict__ sym_b,
                        const float* __restrict__ sym_g, const float* __restrict__ sym_bb,
                        const float* __restrict__ out_w, const float* __restrict__ out_b,
                        const float* __restrict__ norm_g, const float* __restrict__ norm_b,
                        const float* __restrict__ invsig, float* __restrict__ out,
                        int n) {
    __shared__ float hx[16 * HID];  // h_gat tile
    __shared__ float hp[16 * HID];  // predictive-coding branch
    __shared__ float ht[16 * HID];  // mgf intermediate / final pre-LN
    __shared__ float hm[16 * HID];  // mgf branch
    int r0 = blockIdx.x * 16;
    int tid = threadIdx.x, wave = tid >> 5, lane = tid & 31;
    int m = lane & 15;
    int kb = (lane >> 4) << 1;
    int col = wave * 16 + m;
    int rb = (lane >> 4) << 3;
    float is = invsig[0];

    // load h_gat = agg + gat_b
    for (int i = tid; i < 16 * HID; i += 256) {
        int r = i >> 7, c = i & 127;
        int gr = r0 + r;
        hx[i] = (gr < n) ? (agg[(long long)gr * HID + c] + gat_b[c]) : 0.0f;
    }
    __syncthreads();

    // GEMM1: hp = hx @ (pc_w / sigma) + pc_b ; then LN + ReLU
    {
        v8f c = {};
        c = wmma_tile_k128(hx, pc_w, is, m, kb, col, c);
        for (int v = 0; v < 8; v++) hp[(rb + v) * HID + col] = c[v] + pc_b[col];
    }
    __syncthreads();
    ln_rows_inplace(hp, pc_g, pc_bb, wave, lane, true);
    __syncthreads();

    // GEMM2: ht = hx @ mgf_w + mgf_b
    {
        v8f c = {};
        c = wmma_tile_k128(hx, mgf_w, 1.0f, m, kb, col, c);
        for (int v = 0; v < 8; v++) ht[(rb + v) * HID + col] = c[v] + mgf_b[col];
    }
    __syncthreads();

    // GEMM3: hm = ht @ sym_w + sym_b ; then LN + ReLU
    {
        v8f c = {};
        c = wmma_tile_k128(ht, sym_w, 1.0f, m, kb, col, c);
        for (int v = 0; v < 8; v++) hm[(rb + v) * HID + col] = c[v] + sym_b[col];
    }
    __syncthreads();
    ln_rows_inplace(hm, sym_g, sym_bb, wave, lane, true);
    __syncthreads();

    // GEMM4: o = hx@W[0:128] + hp@W[128:256] + hm@W[256:384] + out_b  (K = 384)
    {
        v8f c = {};
        c = wmma_tile_k128(hx, out_w, 1.0f, m, kb, col, c);
        c = wmma_tile_k128(hp, out_w + 128 * HID, 1.0f, m, kb, col, c);
        c = wmma_tile_k128(hm, out_w + 256 * HID, 1.0f, m, kb, col, c);
        for (int v = 0; v < 8; v++) ht[(rb + v) * HID + col] = c[v] + out_b[col];
    }
    __syncthreads();

    // final LayerNorm (no ReLU), write to global
    for (int rr = 0; rr < 2; rr++) {
        int r = wave * 2 + rr;
        int gr = r0 + r;
        float s = 0.0f;
        for (int c = lane; c < HID; c += 32) s += ht[r * HID + c];
        for (int o = 16; o > 0; o >>= 1) s += __shfl_down(s, o, 32);
        s = __shfl(s, 0, 32);
        float mean = s * (1.0f / HID);
        float vs = 0.0f;
        for (int c = lane; c < HID; c += 32) {
            float d = ht[r * HID + c] - mean;
            vs += d * d;
        }
        for (int o = 16; o > 0; o >>= 1) vs += __shfl_down(vs, o, 32);
        vs = __shfl(vs, 0, 32);
        float inv = rsqrtf(vs * (1.0f / HID) + LN_EPS);
        if (gr < n) {
            for (int c = lane; c < HID; c += 32) {
                out[(long long)gr * HID + c] =
                    (ht[r * HID + c] - mean) * inv * norm_g[c] + norm_b[c];
            }
        }
    }
}

// ---------------------------------------------------------------------------
// launch
// ---------------------------------------------------------------------------
extern "C" void kernel_launch(void* const* d_in, const int* in_sizes, int n_in,
                              void* d_out, int out_size, void* d_ws, size_t ws_size,
                              hipStream_t stream) {
    const float* x       = (const float*)d_in[0];
    // d_in[1] edge_weight: unused by reference
    const float* gat_w   = (const float*)d_in[2];
    const float* att_src = (const float*)d_in[3];
    const float* att_dst = (const float*)d_in[4];
    const float* gat_b   = (const float*)d_in[5];
    const float* pc_w    = (const float*)d_in[6];
    const float* pc_b    = (const float*)d_in[7];
    const float* pc_g    = (const float*)d_in[8];
    const float* pc_bb   = (const float*)d_in[9];
    const float* mgf_w   = (const float*)d_in[10];
    const float* mgf_b   = (const float*)d_in[11];
    const float* sym_w   = (const float*)d_in[12];
    const float* sym_b   = (const float*)d_in[13];
    const float* sym_g   = (const float*)d_in[14];
    const float* sym_bb  = (const float*)d_in[15];
    const float* out_w   = (const float*)d_in[16];
    const float* out_b   = (const float*)d_in[17];
    const float* norm_g  = (const float*)d_in[18];
    const float* norm_b  = (const float*)d_in[19];
    const int*   ei      = (const int*)d_in[20];
    // d_in[21] batch: unused by reference

    const int N = in_sizes[0] / HID;
    const int E = in_sizes[20] / 2;
    float* out = (float*)d_out;

    // workspace layout (floats)
    float* ws    = (float*)d_ws;
    float* h0    = ws;                        // N*128
    float* asrc  = h0 + (size_t)N * HID;      // N*4
    float* adst  = asrc + (size_t)N * HEADS;  // N*4
    float* mbuf  = adst + (size_t)N * HEADS;  // N*4
    float* denom = mbuf + (size_t)N * HEADS;  // N*4
    float* agg   = denom + (size_t)N * HEADS; // N*128
    float* eexp  = agg + (size_t)N * HID;     // (E+N)*4
    float* invs  = eexp + (size_t)(E + N) * HEADS;  // 1

    const int tiles = (N + 15) / 16;

    k_gemm_xw<<<tiles, 256, 0, stream>>>(x, gat_w, h0, N);
    k_att_logits<<<(N * HEADS + 255) / 256, 256, 0, stream>>>(h0, att_src, att_dst,
                                                              asrc, adst, N);
    k_init<<<2048, 256, 0, stream>>>(mbuf, denom, agg, N);

    long long edge_threads = (long long)(E + N) * HEADS;
    int eb = (int)((edge_threads + 255) / 256);
    k_edge_max<<<eb, 256, 0, stream>>>(ei, asrc, adst, mbuf, E, N);
    k_edge_exp<<<eb, 256, 0, stream>>>(ei, asrc, adst, mbuf, eexp, denom, E, N);

    long long msg_threads = (long long)(E + N) * 32;
    int mb = (int)((msg_threads + 255) / 256);
    k_message<<<mb, 256, 0, stream>>>(ei, h0, eexp, denom, agg, E, N);

    k_sigma<<<1, HID, 0, stream>>>(pc_w, invs);

    k_fused<<<tiles, 256, 0, stream>>>(agg, gat_b, pc_w, pc_b, pc_g, pc_bb,
                                       mgf_w, mgf_b, sym_w, sym_b, sym_g, sym_bb,
                                       out_w, out_b, norm_g, norm_b, invs, out, N);
}